// DiWeightedGCNLayer_40089224741067
// MI455X (gfx1250) — compile-verified
//
#include <hip/hip_runtime.h>
#include <hip/hip_bf16.h>
#include <math.h>

// ---------------------------------------------------------------------------
// DiWeightedGCNLayer for MI455X (gfx1250, wave32)
//   h   = LayerNorm(x) * gamma + beta
//   m   = h @ W^T + b                      (fp32 WMMA 16x16x4)
//   out = segment_mean_weighted(m[src]*w, dst) * dst_scale
//   y   = x + gelu_exact(out)
// ---------------------------------------------------------------------------

typedef float v2f __attribute__((ext_vector_type(2)));
typedef float v8f __attribute__((ext_vector_type(8)));

#define DIM 128
#define LDS_PITCH 132   // 128 + 4 pad: column reads hit 16 distinct banks

// ---------------------------------------------------------------------------
// Kernel 1: fused LayerNorm + GEMM (m = LN(x) @ W^T + b)
// Block = 256 threads = 8 waves. Block handles 16 rows; wave w computes the
// 16x16 output tile at columns [16w, 16w+16).
// ---------------------------------------------------------------------------
__global__ __launch_bounds__(256) void ln_gemm_kernel(
    const float* __restrict__ x,
    const float* __restrict__ gamma,
    const float* __restrict__ beta,
    const float* __restrict__ W,      // [DIM, DIM] row-major (out, in)
    const float* __restrict__ bias,   // [DIM]
    float* __restrict__ m,            // [N, DIM]
    int N)
{
    __shared__ float hS[16 * LDS_PITCH];
    __shared__ float mu_s[16];
    __shared__ float rs_s[16];

    const int tid = threadIdx.x;
    const int r0  = blockIdx.x * 16;
    const bool full_tile = (r0 + 16 <= N);

    // ---- stage 16x128 x-tile into LDS (16 threads per row, 8 cols each) ----
    {
        const int row = tid >> 4;          // 0..15
        const int c0  = (tid & 15) * 8;    // 0,8,...,120
        if (full_tile || (r0 + row < N)) {
            const float4 v0 = *(const float4*)(x + (size_t)(r0 + row) * DIM + c0);
            const float4 v1 = *(const float4*)(x + (size_t)(r0 + row) * DIM + c0 + 4);
            *(float4*)(&hS[row * LDS_PITCH + c0])     = v0;
            *(float4*)(&hS[row * LDS_PITCH + c0 + 4]) = v1;
        } else {
            *(float4*)(&hS[row * LDS_PITCH + c0])     = float4{0.f, 0.f, 0.f, 0.f};
            *(float4*)(&hS[row * LDS_PITCH + c0 + 4]) = float4{0.f, 0.f, 0.f, 0.f};
        }
    }
    __syncthreads();

    // ---- per-row mean / rsqrt(var + eps): one thread per row ----
    if (tid < 16) {
        const float4* row4 = (const float4*)(&hS[tid * LDS_PITCH]);
        float s = 0.f;
        #pragma unroll
        for (int i = 0; i < 32; ++i) {
            float4 v = row4[i];
            s += v.x + v.y + v.z + v.w;
        }
        const float mu = s * (1.0f / DIM);
        float vs = 0.f;
        #pragma unroll
        for (int i = 0; i < 32; ++i) {
            float4 v = row4[i];
            float a = v.x - mu, b = v.y - mu, c = v.z - mu, d = v.w - mu;
            vs += a * a + b * b + c * c + d * d;
        }
        mu_s[tid] = mu;
        rs_s[tid] = rsqrtf(vs * (1.0f / DIM) + 1e-5f);
    }
    __syncthreads();

    // ---- normalize in place: h = (x - mu) * rs * gamma + beta ----
    {
        const int row = tid >> 4;
        const int c0  = (tid & 15) * 8;
        const float mu = mu_s[row];
        const float rs = rs_s[row];
        #pragma unroll
        for (int q = 0; q < 2; ++q) {
            const int c = c0 + 4 * q;
            float4 v = *(float4*)(&hS[row * LDS_PITCH + c]);
            const float4 g = *(const float4*)(gamma + c);
            const float4 bb = *(const float4*)(beta + c);
            v.x = (v.x - mu) * rs * g.x + bb.x;
            v.y = (v.y - mu) * rs * g.y + bb.y;
            v.z = (v.z - mu) * rs * g.z + bb.z;
            v.w = (v.w - mu) * rs * g.w + bb.w;
            *(float4*)(&hS[row * LDS_PITCH + c]) = v;
        }
    }
    __syncthreads();

    // ---- WMMA: each wave computes one 16x16 tile of m over K = 128 ----
    const int wave = tid >> 5;         // 0..7 -> column tile
    const int lane = tid & 31;
    const int half = lane >> 4;        // 0 | 1
    const int l16  = lane & 15;        // A: row M;  B: col N
    const int j0   = wave * 16;

    v8f acc = {};                      // C = 0
    const float* Wrow = W + (size_t)(j0 + l16) * DIM;  // B[k][n] = W[j0+n][k]

    #pragma unroll 4
    for (int k0 = 0; k0 < DIM; k0 += 4) {
        const int kk = k0 + 2 * half;
        // A 16x4 f32 layout: VGPR0 = K {0|2}, VGPR1 = K {1|3}; M = lane&15
        v2f a;
        a.x = hS[l16 * LDS_PITCH + kk];
        a.y = hS[l16 * LDS_PITCH + kk + 1];
        // B 4x16 f32 layout (mirrored): VGPR0 = K {0|2}, VGPR1 = K {1|3}; N = lane&15
        const float2 wv = *(const float2*)(Wrow + kk);
        v2f b;
        b.x = wv.x;
        b.y = wv.y;
        acc = __builtin_amdgcn_wmma_f32_16x16x4_f32(
            /*neg_a=*/false, a, /*neg_b=*/false, b,
            /*c_mod=*/(short)0, acc, /*reuse_a=*/false, /*reuse_b=*/false);
    }

    // ---- bias + store: C/D layout: VGPR v -> row (v + 8*half), col l16 ----
    const float bn = bias[j0 + l16];
    float* mg = m + (size_t)r0 * DIM + j0 + l16;
    if (full_tile) {
        // fast path: no per-row guards -> no exec-mask churn
        #pragma unroll
        for (int v = 0; v < 8; ++v) {
            const int row = v + 8 * half;
            mg[(size_t)row * DIM] = acc[v] + bn;
        }
    } else {
        #pragma unroll
        for (int v = 0; v < 8; ++v) {
            const int row = v + 8 * half;
            if (r0 + row < N) {
                mg[(size_t)row * DIM] = acc[v] + bn;
            }
        }
    }
}

// ---------------------------------------------------------------------------
// Kernel 2: zero the accumulator + degree region of workspace
// ---------------------------------------------------------------------------
__global__ __launch_bounds__(256) void zero_kernel(float* __restrict__ p, long long n)
{
    const long long i = (long long)blockIdx.x * 256 + threadIdx.x;
    if (i < n) p[i] = 0.0f;
}

// ---------------------------------------------------------------------------
// Kernel 3: edge scatter.  One wave per edge: gather m[src] (float4/lane),
// weight, atomic-add into acc[dst]; lane 0 accumulates weighted degree.
// ---------------------------------------------------------------------------
__global__ __launch_bounds__(256) void scatter_kernel(
    const long long* __restrict__ edge_index,  // [2, E] int64
    const float* __restrict__ edge_weight,     // [E]
    const float* __restrict__ m,               // [N, DIM]
    float* __restrict__ acc,                   // [N, DIM]
    float* __restrict__ deg,                   // [N]
    long long E)
{
    const long long e = (long long)blockIdx.x * 8 + (threadIdx.x >> 5);
    if (e >= E) return;
    const int lane = threadIdx.x & 31;

    const long long s = edge_index[e];       // wave-uniform -> scalarized
    const long long d = edge_index[E + e];
    const float w = edge_weight[e];

    const float4 v = *(const float4*)(m + s * DIM + lane * 4);
    float* o = acc + d * DIM + lane * 4;
    atomicAdd(o + 0, v.x * w);
    atomicAdd(o + 1, v.y * w);
    atomicAdd(o + 2, v.z * w);
    atomicAdd(o + 3, v.w * w);
    if (lane == 0) atomicAdd(deg + d, w);
}

// ---------------------------------------------------------------------------
// Kernel 4: finalize  y = x + gelu_exact(acc / max(deg,1) * dst_scale)
// ---------------------------------------------------------------------------
__global__ __launch_bounds__(256) void finalize_kernel(
    const float* __restrict__ x,
    const float* __restrict__ acc,
    const float* __restrict__ deg,
    const float* __restrict__ dst_scale,
    float* __restrict__ y,
    long long total)   // N * DIM
{
    const long long i = (long long)blockIdx.x * 256 + threadIdx.x;
    if (i >= total) return;
    const long long row = i >> 7;   // / DIM
    const float dg = fmaxf(deg[row], 1.0f);
    const float o = (acc[i] / dg) * dst_scale[row];
    // exact GELU: 0.5*o*(1+erf(o/sqrt(2)))
    const float g = 0.5f * o * (1.0f + erff(o * 0.70710678118654752f));
    y[i] = x[i] + g;
}

// ---------------------------------------------------------------------------
// Launch
// ---------------------------------------------------------------------------
extern "C" void kernel_launch(void* const* d_in, const int* in_sizes, int n_in,
                              void* d_out, int out_size, void* d_ws, size_t ws_size,
                              hipStream_t stream) {
    // setup_inputs order:
    // 0:x [N,128] f32 | 1:gamma [128] | 2:beta [128] | 3:W [128,128] | 4:b [128]
    // 5:edge_index [2,E] i64 | 6:num_nodes (scalar) | 7:edge_weight [E] | 8:dst_scale [N]
    const float*     x   = (const float*)d_in[0];
    const float*     gam = (const float*)d_in[1];
    const float*     bet = (const float*)d_in[2];
    const float*     W   = (const float*)d_in[3];
    const float*     bia = (const float*)d_in[4];
    const long long* ei  = (const long long*)d_in[5];
    const float*     ew  = (const float*)d_in[7];
    const float*     dsc = (const float*)d_in[8];

    const long long N = (long long)in_sizes[0] / DIM;
    const long long E = (long long)in_sizes[7];

    // workspace layout: m[N*DIM] | acc[N*DIM] | deg[N]   (fp32)
    float* m_buf = (float*)d_ws;
    float* acc   = m_buf + (size_t)N * DIM;
    float* deg   = acc + (size_t)N * DIM;
    float* y     = (float*)d_out;

    // 1) fused LayerNorm + GEMM via fp32 WMMA
    {
        dim3 grid((unsigned)((N + 15) / 16));
        ln_gemm_kernel<<<grid, 256, 0, stream>>>(x, gam, bet, W, bia, m_buf, (int)N);
    }
    // 2) clear acc + deg (contiguous N*DIM + N floats)
    {
        const long long n = N * DIM + N;
        dim3 grid((unsigned)((n + 255) / 256));
        zero_kernel<<<grid, 256, 0, stream>>>(acc, n);
    }
    // 3) edge gather + weighted scatter-add
    {
        dim3 grid((unsigned)((E + 7) / 8));
        scatter_kernel<<<grid, 256, 0, stream>>>(ei, ew, m_buf, acc, deg, E);
    }
    // 4) normalize by degree, scale, residual + exact GELU
    {
        const long long total = N * DIM;
        dim3 grid((unsigned)((total + 255) / 256));
        finalize_kernel<<<grid, 256, 0, stream>>>(x, acc, deg, dsc, y, total);
    }
}